// YoloLayer_86818468922218
// MI455X (gfx1250) — compile-verified
//
#include <hip/hip_runtime.h>

// YOLO layer decode for MI455X (gfx1250, wave32).
// Memory-bound layout transform + elementwise math:
//   ~88.7 MB traffic / 23.3 TB/s  =>  ~3.8 us floor. No GEMM => no WMMA.
// Strategy: coalesced b32 loads (imm-offset channel strides), LDS-staged
// transpose, then CDNA5 GLOBAL_STORE_ASYNC_FROM_LDS_B128 to DMA the staged
// tile straight from LDS to memory as aligned 16B/lane full-cacheline writes
// (ASYNCcnt path; falls back to ds_load+global_store if builtin is absent).

namespace {
constexpr int B_   = 64;
constexpr int NA_  = 3;
constexpr int NC_  = 3;
constexpr int G_   = 76;
constexpr int CH   = NC_ + 7;              // 10 channels per anchor
constexpr int GG   = G_ * G_;              // 5776
constexpr float STRIDE_ = 608.0f / G_;     // 8.0
constexpr int NBOX = B_ * NA_ * GG;        // 1,108,992 (exactly 4332 * 256)
constexpr int TPB  = 256;                  // 8 wave32s per block
constexpr int FLOATS_PER_BLOCK = TPB * CH; // 2560 floats = 10 KiB
constexpr int VEC4_PER_BLOCK   = FLOATS_PER_BLOCK / 4; // 640
}

#if defined(__has_builtin)
#if __has_builtin(__builtin_amdgcn_global_store_async_from_lds_b128)
#define USE_ASYNC_LDS_STORE 1
#endif
#endif
#ifndef USE_ASYNC_LDS_STORE
#define USE_ASYNC_LDS_STORE 0
#endif

#if USE_ASYNC_LDS_STORE
// Builtin signature (from the round-2 diagnostic): pointee is a 16-byte int
// vector; global side is addrspace(1), LDS side is addrspace(3).
typedef int v4i_t __attribute__((vector_size(4 * sizeof(int))));
typedef __attribute__((address_space(1))) v4i_t* gv4i_p;
typedef __attribute__((address_space(3))) v4i_t* lv4i_p;
#endif

__device__ __forceinline__ float sigmoidf_(float v) {
    return 1.0f / (1.0f + expf(-v));
}

__global__ void __launch_bounds__(TPB)
yolo_layer_kernel(const float* __restrict__ x,
                  const float* __restrict__ anchors,
                  float* __restrict__ out) {
    const int tid = threadIdx.x;
    const int t   = blockIdx.x * TPB + tid;   // box id; grid is exact, no tail

    // t = (b*NA + a)*GG + s,  s = gy*G + gx
    const int s  = t % GG;
    const int ba = t / GG;                    // b*NA + a
    const int a  = ba % NA_;
    const int gy = s / G_;
    const int gx = s - gy * G_;

    // 10 coalesced channel loads: single address + immediate offsets c*GG*4B.
    const float* __restrict__ base = x + (size_t)ba * (CH * GG) + s;
    float v[CH];
#pragma unroll
    for (int c = 0; c < CH; ++c) v[c] = base[c * GG];

    const float aw = anchors[a * 4 + 0];
    const float ah = anchors[a * 4 + 1];

    // Decode. bw = exp(w) * (anchor_w/STRIDE) * STRIDE = exp(w) * anchor_w.
    float r[CH];
    r[0] = (sigmoidf_(v[0]) + (float)gx) * STRIDE_;  // bx
    r[1] = (sigmoidf_(v[1]) + (float)gy) * STRIDE_;  // by
    r[2] = expf(v[2]) * aw;                          // bw
    r[3] = expf(v[3]) * ah;                          // bh
    r[4] = v[4];                                     // im (raw)
    r[5] = v[5];                                     // re (raw)
    r[6] = sigmoidf_(v[6]);                          // conf
    r[7] = sigmoidf_(v[7]);                          // cls0
    r[8] = sigmoidf_(v[8]);                          // cls1
    r[9] = sigmoidf_(v[9]);                          // cls2

    // Stage in LDS so global stores are aligned b128 full-cacheline writes.
    // Write pattern lds[tid*10 + c]: stride-10 words over 64 banks is
    // conflict-free (gcd(10,64)=2; 5 invertible mod 32).
    __shared__ __align__(16) float lds[FLOATS_PER_BLOCK];
#pragma unroll
    for (int c = 0; c < CH; ++c) lds[tid * CH + c] = r[c];
    __syncthreads();

    // Block output region: 2560 contiguous floats, 10240-byte aligned.
    float4* __restrict__ dst =
        (float4*)(out + (size_t)blockIdx.x * FLOATS_PER_BLOCK);
    float4* src = (float4*)lds;

#if USE_ASYNC_LDS_STORE
    // CDNA5 async DMA: LDS -> global, no VGPR round-trip. Tracked by ASYNCcnt.
#pragma unroll
    for (int p = 0; p < 2; ++p)
        __builtin_amdgcn_global_store_async_from_lds_b128(
            (gv4i_p)(dst + p * TPB + tid),
            (lv4i_p)(src + p * TPB + tid),
            /*imm offset*/ 0, /*cpol*/ 0);
    if (tid < VEC4_PER_BLOCK - 2 * TPB)  // remaining 128 float4s
        __builtin_amdgcn_global_store_async_from_lds_b128(
            (gv4i_p)(dst + 2 * TPB + tid),
            (lv4i_p)(src + 2 * TPB + tid),
            /*imm offset*/ 0, /*cpol*/ 0);
#if __has_builtin(__builtin_amdgcn_s_wait_asynccnt)
    __builtin_amdgcn_s_wait_asynccnt(0);  // S_ENDPGM would also wait-idle
#endif
#else
    // Fallback: ds_load_b128 + global_store_b128 copy.
#pragma unroll
    for (int p = 0; p < 2; ++p)
        dst[p * TPB + tid] = src[p * TPB + tid];
    if (tid < VEC4_PER_BLOCK - 2 * TPB)  // remaining 128 float4s
        dst[2 * TPB + tid] = src[2 * TPB + tid];
#endif
}

extern "C" void kernel_launch(void* const* d_in, const int* in_sizes, int n_in,
                              void* d_out, int out_size, void* d_ws, size_t ws_size,
                              hipStream_t stream) {
    (void)in_sizes; (void)n_in; (void)out_size; (void)d_ws; (void)ws_size;
    const float* x       = (const float*)d_in[0];
    const float* anchors = (const float*)d_in[1];
    float* out           = (float*)d_out;

    static_assert(NBOX % TPB == 0, "exact grid");
    yolo_layer_kernel<<<NBOX / TPB, TPB, 0, stream>>>(x, anchors, out);
}